// AE_AtlasNet_7481833029680
// MI455X (gfx1250) — compile-verified
//
#include <hip/hip_runtime.h>

typedef __attribute__((ext_vector_type(2))) float v2f;
typedef __attribute__((ext_vector_type(8))) float v8f;

#define NB 8L
#define NP 2048L
#define KK 20L
#define PP 2048L
#define BN (NB*NP)          // 16384
#define BNK (BN*KK)         // 327680

// ---------------- workspace layout (float offsets) ----------------
#define OFF_PTS   0L
#define OFF_X1    (OFF_PTS + NB*NP*3)          // 49152
#define OFF_X2    (OFF_X1 + BN*64)
#define OFF_X3    (OFF_X2 + BN*64)
#define OFF_X4    (OFF_X3 + BN*64)
#define OFF_CAT   (OFF_X4 + BN*128)
#define OFF_IDX   (OFF_CAT + BN*320)           // int region, BNK entries
#define OFF_SCALE (OFF_IDX + BNK)
#define OFF_SHIFT (OFF_SCALE + 1056)
#define OFF_GLOB  (OFF_SHIFT + 1056)           // 8192
#define OFF_ZL    (OFF_GLOB + 8192)
#define OFF_LAT   (OFF_ZL + 8192)
#define OFF_R1    (OFF_LAT + 8192)             // big region 1 (EF / Y / HD)
#define R1_SZ     41943040L
#define OFF_R2    (OFF_R1 + R1_SZ)             // big region 2 (Z / Z5 / ZD)

#define WMMA_F32(a, b, c) \
  __builtin_amdgcn_wmma_f32_16x16x4_f32(false, (a), false, (b), (short)0, (c), false, false)

// ---------------- transpose x[B,3,N] -> pts[B*N,3] ----------------
__global__ __launch_bounds__(256) void transpose_kernel(const float* __restrict__ x,
                                                        float* __restrict__ pts) {
  long i = (long)blockIdx.x * 256 + threadIdx.x;
  if (i >= NB * NP * 3) return;
  int c = (int)(i % 3);
  long bn = i / 3;
  long b = bn / NP, n = bn % NP;
  pts[i] = x[(b * 3 + c) * NP + n];
}

// ---------------- kNN: one block per (b,n), top-K by adj value ----------------
__global__ __launch_bounds__(256) void knn_kernel(const float* __restrict__ h, int D,
                                                  int* __restrict__ idx) {
  long bn = blockIdx.x;                 // b*N + n
  long b = bn / NP, n = bn % NP;
  const float* hb = h + b * NP * D;
  __shared__ float hn[64];
  __shared__ float dist[2048];
  __shared__ float rv[256];
  __shared__ int   ri[256];
  int tid = threadIdx.x;
  if (tid < D) hn[tid] = hb[n * (long)D + tid];
  __syncthreads();
  for (int m = tid; m < (int)NP; m += 256) {
    const float* hm = hb + (long)m * D;
    float dot = 0.f, nrm = 0.f;
    for (int c = 0; c < D; ++c) { float v = hm[c]; dot += hn[c] * v; nrm += v * v; }
    dist[m] = 2.f * dot - nrm;          // constant -|hn|^2 shift dropped (rank-invariant)
  }
  __syncthreads();
  for (int j = 0; j < (int)KK; ++j) {
    float best = -__builtin_inff(); int bi = 0x7fffffff;
    for (int m = tid; m < (int)NP; m += 256) {
      float v = dist[m];
      if (v > best) { best = v; bi = m; }     // strict > keeps smallest m in-thread
    }
    rv[tid] = best; ri[tid] = bi;
    __syncthreads();
    for (int s = 128; s > 0; s >>= 1) {
      if (tid < s) {
        if (rv[tid + s] > rv[tid] ||
            (rv[tid + s] == rv[tid] && ri[tid + s] < ri[tid])) {
          rv[tid] = rv[tid + s]; ri[tid] = ri[tid + s];
        }
      }
      __syncthreads();
    }
    if (tid == 0) { idx[bn * KK + j] = ri[0]; dist[ri[0]] = -__builtin_inff(); }
    __syncthreads();
  }
}

// ---------------- edge features EF[B*N*K, 2D] ----------------
__global__ __launch_bounds__(256) void ef_kernel(const float* __restrict__ h,
                                                 const int* __restrict__ idx, int D,
                                                 float* __restrict__ EF, long total) {
  long i = (long)blockIdx.x * 256 + threadIdx.x;
  if (i >= total) return;
  int C2 = 2 * D;
  long r = i / C2; int c = (int)(i % C2);     // r = bn*K + k
  long bn = r / KK;
  float v;
  if (c < D) v = h[bn * D + c];
  else {
    int mi = idx[r];
    long b = bn / NP;
    v = h[(b * NP + mi) * D + (c - D)] - h[bn * D + (c - D)];
  }
  EF[i] = v;
}

// ---------------- WMMA f32 GEMM: out[r,o] = sum_c A[r,c]*W[o,c] + bias[o] ----------------
// One wave computes a 16x64 output strip: 4 accumulators share each A fragment.
// Fast path: K unrolled by 8 (two independent fragment sets -> 8 WMMAs per
// iteration, loads of both sets hoisted ahead of the first WMMA group);
// fragments loaded as v2f (global_load_b64) when C is even.
__global__ __launch_bounds__(256) void gemm_wmma(const float* __restrict__ A,
                                                 const float* __restrict__ W,
                                                 const float* __restrict__ bias,
                                                 float* __restrict__ out,
                                                 int M, int C, int O, int relu) {
  int wave = (int)(((long)blockIdx.x * blockDim.x + threadIdx.x) >> 5);
  int lane = threadIdx.x & 31;
  int colGroups = (O + 63) >> 6;
  int rowTiles = (M + 15) >> 4;
  if (wave >= rowTiles * colGroups) return;
  int tr = wave / colGroups, tg = wave % colGroups;
  int l15 = lane & 15;
  int chalf = (lane >> 4) * 2;                 // lanes 16-31 hold K+2,K+3
  int row = tr * 16 + l15;
  int ocBase = tg * 64 + l15;                  // this lane's col within each of 4 tiles
  int C4 = C & ~3;
  v8f acc[4] = {};
  bool full = (tr * 16 + 15 < M) && (tg * 64 + 63 < O);

  if (full) {
    const float* pa  = A + (size_t)row * C + chalf;
    const float* pb0 = W + (size_t)ocBase * C + chalf;
    const float* pb1 = pb0 + (size_t)16 * C;
    const float* pb2 = pb0 + (size_t)32 * C;
    const float* pb3 = pb0 + (size_t)48 * C;
    int c0 = 0;
    if ((C & 1) == 0) {
      // 8-byte aligned: use v2f (b64) fragment loads
      const v2f* qa  = (const v2f*)pa;
      const v2f* qb0 = (const v2f*)pb0;
      const v2f* qb1 = (const v2f*)pb1;
      const v2f* qb2 = (const v2f*)pb2;
      const v2f* qb3 = (const v2f*)pb3;
      for (; c0 + 8 <= C4; c0 += 8) {
        int i0 = c0 >> 1, i1 = i0 + 2;
        v2f a0 = qa[i0], a1 = qa[i1];
        v2f b00 = qb0[i0], b01 = qb1[i0], b02 = qb2[i0], b03 = qb3[i0];
        v2f b10 = qb0[i1], b11 = qb1[i1], b12 = qb2[i1], b13 = qb3[i1];
        acc[0] = WMMA_F32(a0, b00, acc[0]);
        acc[1] = WMMA_F32(a0, b01, acc[1]);
        acc[2] = WMMA_F32(a0, b02, acc[2]);
        acc[3] = WMMA_F32(a0, b03, acc[3]);
        acc[0] = WMMA_F32(a1, b10, acc[0]);
        acc[1] = WMMA_F32(a1, b11, acc[1]);
        acc[2] = WMMA_F32(a1, b12, acc[2]);
        acc[3] = WMMA_F32(a1, b13, acc[3]);
      }
      for (; c0 < C4; c0 += 4) {
        int i0 = c0 >> 1;
        v2f a0 = qa[i0];
        v2f b00 = qb0[i0], b01 = qb1[i0], b02 = qb2[i0], b03 = qb3[i0];
        acc[0] = WMMA_F32(a0, b00, acc[0]);
        acc[1] = WMMA_F32(a0, b01, acc[1]);
        acc[2] = WMMA_F32(a0, b02, acc[2]);
        acc[3] = WMMA_F32(a0, b03, acc[3]);
      }
    } else {
      // odd C: scalar-pair fragment loads
      for (; c0 + 8 <= C4; c0 += 8) {
        int i1 = c0 + 4;
        v2f a0, a1, b00, b01, b02, b03, b10, b11, b12, b13;
        a0.x  = pa[c0];      a0.y  = pa[c0 + 1];
        a1.x  = pa[i1];      a1.y  = pa[i1 + 1];
        b00.x = pb0[c0];     b00.y = pb0[c0 + 1];
        b01.x = pb1[c0];     b01.y = pb1[c0 + 1];
        b02.x = pb2[c0];     b02.y = pb2[c0 + 1];
        b03.x = pb3[c0];     b03.y = pb3[c0 + 1];
        b10.x = pb0[i1];     b10.y = pb0[i1 + 1];
        b11.x = pb1[i1];     b11.y = pb1[i1 + 1];
        b12.x = pb2[i1];     b12.y = pb2[i1 + 1];
        b13.x = pb3[i1];     b13.y = pb3[i1 + 1];
        acc[0] = WMMA_F32(a0, b00, acc[0]);
        acc[1] = WMMA_F32(a0, b01, acc[1]);
        acc[2] = WMMA_F32(a0, b02, acc[2]);
        acc[3] = WMMA_F32(a0, b03, acc[3]);
        acc[0] = WMMA_F32(a1, b10, acc[0]);
        acc[1] = WMMA_F32(a1, b11, acc[1]);
        acc[2] = WMMA_F32(a1, b12, acc[2]);
        acc[3] = WMMA_F32(a1, b13, acc[3]);
      }
      for (; c0 < C4; c0 += 4) {
        v2f a0, b00, b01, b02, b03;
        a0.x  = pa[c0];  a0.y  = pa[c0 + 1];
        b00.x = pb0[c0]; b00.y = pb0[c0 + 1];
        b01.x = pb1[c0]; b01.y = pb1[c0 + 1];
        b02.x = pb2[c0]; b02.y = pb2[c0 + 1];
        b03.x = pb3[c0]; b03.y = pb3[c0 + 1];
        acc[0] = WMMA_F32(a0, b00, acc[0]);
        acc[1] = WMMA_F32(a0, b01, acc[1]);
        acc[2] = WMMA_F32(a0, b02, acc[2]);
        acc[3] = WMMA_F32(a0, b03, acc[3]);
      }
    }
    if (C4 < C) {                              // ragged C tail (C%4 != 0), guard on c only
      bool va = (C4 + chalf) < C, vb = (C4 + chalf + 1) < C;
      v2f at, t0, t1, t2, t3;
      at.x = va ? pa[C4]  : 0.f; at.y = vb ? pa[C4 + 1]  : 0.f;
      t0.x = va ? pb0[C4] : 0.f; t0.y = vb ? pb0[C4 + 1] : 0.f;
      t1.x = va ? pb1[C4] : 0.f; t1.y = vb ? pb1[C4 + 1] : 0.f;
      t2.x = va ? pb2[C4] : 0.f; t2.y = vb ? pb2[C4 + 1] : 0.f;
      t3.x = va ? pb3[C4] : 0.f; t3.y = vb ? pb3[C4 + 1] : 0.f;
      acc[0] = WMMA_F32(at, t0, acc[0]);
      acc[1] = WMMA_F32(at, t1, acc[1]);
      acc[2] = WMMA_F32(at, t2, acc[2]);
      acc[3] = WMMA_F32(at, t3, acc[3]);
    }
  } else {
    for (int c0 = 0; c0 < C; c0 += 4) {
      int ca = c0 + chalf, cb = ca + 1;
      bool ra = row < M;
      bool va = ca < C, vb = cb < C;
      v2f a, b[4];
      a.x = (ra && va) ? A[(size_t)row * C + ca] : 0.f;
      a.y = (ra && vb) ? A[(size_t)row * C + cb] : 0.f;
#pragma unroll
      for (int t = 0; t < 4; ++t) {
        int oc = ocBase + t * 16;
        bool vo = oc < O;
        b[t].x = (vo && va) ? W[(size_t)oc * C + ca] : 0.f;
        b[t].y = (vo && vb) ? W[(size_t)oc * C + cb] : 0.f;
      }
      acc[0] = WMMA_F32(a, b[0], acc[0]);
      acc[1] = WMMA_F32(a, b[1], acc[1]);
      acc[2] = WMMA_F32(a, b[2], acc[2]);
      acc[3] = WMMA_F32(a, b[3], acc[3]);
    }
  }

  // epilogue: D layout: vgpr j -> (M = tr*16 + j + (lane>=16)*8, N = lane&15)
  int rbase = tr * 16 + (lane >> 4) * 8;
  if (full) {
#pragma unroll
    for (int t = 0; t < 4; ++t) {
      int oc = ocBase + t * 16;
      float bv = bias[oc];
#pragma unroll
      for (int j = 0; j < 8; ++j) {
        float v = acc[t][j] + bv;
        if (relu) v = fmaxf(v, 0.f);
        out[(size_t)(rbase + j) * O + oc] = v;
      }
    }
  } else {
#pragma unroll
    for (int t = 0; t < 4; ++t) {
      int oc = ocBase + t * 16;
      if (oc < O) {
        float bv = bias[oc];
#pragma unroll
        for (int j = 0; j < 8; ++j) {
          int r = rbase + j;
          if (r < M) {
            float v = acc[t][j] + bv;
            if (relu) v = fmaxf(v, 0.f);
            out[(size_t)r * O + oc] = v;
          }
        }
      }
    }
  }
}

// ---------------- per-channel BN stats -> scale/shift (one block per channel) ----------------
__global__ __launch_bounds__(256) void stats_kernel(const float* __restrict__ Z, long R, int O,
                                                    const float* __restrict__ g,
                                                    const float* __restrict__ be,
                                                    float* __restrict__ scale,
                                                    float* __restrict__ shift) {
  int o = blockIdx.x, tid = threadIdx.x;
  __shared__ float s1[256], s2[256];
  float a = 0.f, b = 0.f;
  for (long r = tid; r < R; r += 256) {
    float v = Z[r * O + o];
    a += v; b += v * v;
  }
  s1[tid] = a; s2[tid] = b;
  __syncthreads();
  for (int s = 128; s > 0; s >>= 1) {
    if (tid < s) { s1[tid] += s1[tid + s]; s2[tid] += s2[tid + s]; }
    __syncthreads();
  }
  if (tid == 0) {
    float mean = s1[0] / (float)R;
    float var  = s2[0] / (float)R - mean * mean;
    float sc = g[o] * rsqrtf(var + 1e-5f);
    scale[o] = sc;
    shift[o] = be[o] - mean * sc;
  }
}

// ---------------- bn then max over group of S rows: out[g,o] ----------------
__global__ __launch_bounds__(256) void bnmax_kernel(const float* __restrict__ Z,
                                                    const float* __restrict__ scale,
                                                    const float* __restrict__ shift,
                                                    long G, int S, int O,
                                                    float* __restrict__ out) {
  long i = (long)blockIdx.x * 256 + threadIdx.x;
  if (i >= G * O) return;
  long g = i / O; int o = (int)(i % O);
  float sc = scale[o], sh = shift[o];
  const float* p = Z + g * (long)S * O + o;
  float m = -__builtin_inff();
  for (int s = 0; s < S; ++s) m = fmaxf(m, p[(long)s * O] * sc + sh);
  out[i] = m;
}

// ---------------- elementwise bn (+relu) ----------------
__global__ __launch_bounds__(256) void bnrelu_kernel(const float* __restrict__ Z,
                                                     const float* __restrict__ scale,
                                                     const float* __restrict__ shift,
                                                     long total, int O,
                                                     float* __restrict__ out) {
  long i = (long)blockIdx.x * 256 + threadIdx.x;
  if (i >= total) return;
  int o = (int)(i % O);
  out[i] = fmaxf(Z[i] * scale[o] + shift[o], 0.f);
}

// ---------------- concat x1..x4 -> cat[B*N,320] ----------------
__global__ __launch_bounds__(256) void cat_kernel(const float* __restrict__ x1,
                                                  const float* __restrict__ x2,
                                                  const float* __restrict__ x3,
                                                  const float* __restrict__ x4,
                                                  float* __restrict__ cat) {
  long i = (long)blockIdx.x * 256 + threadIdx.x;
  if (i >= BN * 320) return;
  int c = (int)(i % 320); long r = i / 320;
  float v;
  if (c < 64)       v = x1[r * 64 + c];
  else if (c < 128) v = x2[r * 64 + (c - 64)];
  else if (c < 192) v = x3[r * 64 + (c - 128)];
  else              v = x4[r * 128 + (c - 192)];
  cat[i] = v;
}

// ---------------- build decoder input Y[B*P, 1026] ----------------
__global__ __launch_bounds__(256) void buildY_kernel(const float* __restrict__ rg,
                                                     const float* __restrict__ lat,
                                                     float* __restrict__ Y) {
  long i = (long)blockIdx.x * 256 + threadIdx.x;
  if (i >= NB * PP * 1026) return;
  int c = (int)(i % 1026); long r = i / 1026;
  long b = r / PP, l = r % PP;
  Y[i] = (c < 2) ? rg[(b * 2 + c) * PP + l] : lat[b * 1024 + (c - 2)];
}

// ---------------- final conv (O=3) + tanh -> out[B,3,P] ----------------
__global__ __launch_bounds__(256) void final_kernel(const float* __restrict__ H,
                                                    const float* __restrict__ W,
                                                    const float* __restrict__ bias,
                                                    float* __restrict__ out) {
  long i = (long)blockIdx.x * 256 + threadIdx.x;
  if (i >= NB * PP) return;
  long b = i / PP, l = i % PP;
  const float* h = H + i * 256;
  for (int o = 0; o < 3; ++o) {
    float s = bias[o];
    const float* w = W + o * 256;
    for (int c = 0; c < 256; ++c) s += w[c] * h[c];
    out[(b * 3 + o) * PP + l] = tanhf(s);
  }
}

static inline long cdivl(long a, long b) { return (a + b - 1) / b; }

static void launch_gemm(const float* A, const float* W, const float* bias, float* out,
                        long M, int C, int O, int relu, hipStream_t stream) {
  long waves = cdivl(M, 16) * cdivl(O, 64);
  long blocks = cdivl(waves * 32, 256);
  gemm_wmma<<<(int)blocks, 256, 0, stream>>>(A, W, bias, out, (int)M, C, O, relu);
}

extern "C" void kernel_launch(void* const* d_in, const int* in_sizes, int n_in,
                              void* d_out, int out_size, void* d_ws, size_t ws_size,
                              hipStream_t stream) {
  const float* x  = (const float*)d_in[0];
  const float* rg = (const float*)d_in[1];
  const float *Wenc[5], *benc[5], *genc[5], *beenc[5];
  for (int i = 0; i < 5; ++i) {
    Wenc[i]  = (const float*)d_in[2 + 4 * i + 0];
    benc[i]  = (const float*)d_in[2 + 4 * i + 1];
    genc[i]  = (const float*)d_in[2 + 4 * i + 2];
    beenc[i] = (const float*)d_in[2 + 4 * i + 3];
  }
  const float* lW  = (const float*)d_in[22];
  const float* lb  = (const float*)d_in[23];
  const float* lg  = (const float*)d_in[24];
  const float* lbe = (const float*)d_in[25];
  const float* dW1 = (const float*)d_in[26]; const float* db1 = (const float*)d_in[27];
  const float* dg1 = (const float*)d_in[28]; const float* dbe1 = (const float*)d_in[29];
  const float* dW2 = (const float*)d_in[30]; const float* db2 = (const float*)d_in[31];
  const float* dg2 = (const float*)d_in[32]; const float* dbe2 = (const float*)d_in[33];
  const float* dW3 = (const float*)d_in[34]; const float* db3 = (const float*)d_in[35];
  const float* dg3 = (const float*)d_in[36]; const float* dbe3 = (const float*)d_in[37];
  const float* dW4 = (const float*)d_in[38]; const float* db4 = (const float*)d_in[39];

  float* w = (float*)d_ws;
  float* pts   = w + OFF_PTS;
  float* x1    = w + OFF_X1;
  float* x2    = w + OFF_X2;
  float* x3    = w + OFF_X3;
  float* x4    = w + OFF_X4;
  float* cat   = w + OFF_CAT;
  int*   idx   = (int*)(w + OFF_IDX);
  float* scale = w + OFF_SCALE;
  float* shift = w + OFF_SHIFT;
  float* glob  = w + OFF_GLOB;
  float* Zl    = w + OFF_ZL;
  float* lat   = w + OFF_LAT;
  float* R1    = w + OFF_R1;   // EF / Y / HD
  float* R2    = w + OFF_R2;   // Z / Z5 / ZD

  // pts = transpose(x)
  transpose_kernel<<<(int)cdivl(NB * NP * 3, 256), 256, 0, stream>>>(x, pts);

  // ---- edge conv layers ----
  const float* hin[4] = {pts, x1, x2, x3};
  int Din[4] = {3, 64, 64, 64};
  int Oout[4] = {64, 64, 64, 128};
  float* xout[4] = {x1, x2, x3, x4};
  for (int l = 0; l < 4; ++l) {
    int D = Din[l], O = Oout[l], C2 = 2 * D;
    knn_kernel<<<(int)BN, 256, 0, stream>>>(hin[l], D, idx);
    long efcnt = BNK * C2;
    ef_kernel<<<(int)cdivl(efcnt, 256), 256, 0, stream>>>(hin[l], idx, D, R1, efcnt);
    launch_gemm(R1, Wenc[l], benc[l], R2, BNK, C2, O, 1, stream);
    stats_kernel<<<O, 256, 0, stream>>>(R2, BNK, O, genc[l], beenc[l], scale, shift);
    bnmax_kernel<<<(int)cdivl(BN * O, 256), 256, 0, stream>>>(R2, scale, shift, BN, (int)KK, O, xout[l]);
  }

  // ---- cat + W5 + global max ----
  cat_kernel<<<(int)cdivl(BN * 320, 256), 256, 0, stream>>>(x1, x2, x3, x4, cat);
  launch_gemm(cat, Wenc[4], benc[4], R2, BN, 320, 1024, 1, stream);
  stats_kernel<<<1024, 256, 0, stream>>>(R2, BN, 1024, genc[4], beenc[4], scale, shift);
  bnmax_kernel<<<(int)cdivl(NB * 1024, 256), 256, 0, stream>>>(R2, scale, shift, NB, (int)NP, 1024, glob);

  // ---- latent: linear -> bn -> relu ----
  launch_gemm(glob, lW, lb, Zl, NB, 1024, 1024, 0, stream);
  stats_kernel<<<1024, 256, 0, stream>>>(Zl, NB, 1024, lg, lbe, scale, shift);
  bnrelu_kernel<<<(int)cdivl(NB * 1024, 256), 256, 0, stream>>>(Zl, scale, shift, NB * 1024, 1024, lat);

  // ---- decoder ----
  float* Y   = R1;                    // [B*P, 1026]
  float* HD1 = R1 + 17000000L;        // [B*P, 1026]
  float* HD2 = R1;                    // [B*P, 513]  (Y dead)
  float* HD3 = R1 + 17000000L;        // [B*P, 256]  (HD1 dead)
  long BP = NB * PP;

  buildY_kernel<<<(int)cdivl(BP * 1026, 256), 256, 0, stream>>>(rg, lat, Y);

  launch_gemm(Y, dW1, db1, R2, BP, 1026, 1026, 0, stream);
  stats_kernel<<<1026, 256, 0, stream>>>(R2, BP, 1026, dg1, dbe1, scale, shift);
  bnrelu_kernel<<<(int)cdivl(BP * 1026, 256), 256, 0, stream>>>(R2, scale, shift, BP * 1026, 1026, HD1);

  launch_gemm(HD1, dW2, db2, R2, BP, 1026, 513, 0, stream);
  stats_kernel<<<513, 256, 0, stream>>>(R2, BP, 513, dg2, dbe2, scale, shift);
  bnrelu_kernel<<<(int)cdivl(BP * 513, 256), 256, 0, stream>>>(R2, scale, shift, BP * 513, 513, HD2);

  launch_gemm(HD2, dW3, db3, R2, BP, 513, 256, 0, stream);
  stats_kernel<<<256, 256, 0, stream>>>(R2, BP, 256, dg3, dbe3, scale, shift);
  bnrelu_kernel<<<(int)cdivl(BP * 256, 256), 256, 0, stream>>>(R2, scale, shift, BP * 256, 256, HD3);

  final_kernel<<<(int)cdivl(BP, 256), 256, 0, stream>>>(HD3, dW4, db4, (float*)d_out);
}